// SelfCorrelationComputation_67714454389070
// MI455X (gfx1250) — compile-verified
//
#include <hip/hip_runtime.h>
#include <stdint.h>

// Problem constants (from reference: q is (1, 32, 24, 24, 24) fp32, K=5, P=2)
#define DCUBE 24           // l == w == h == 24
#define PD    28           // padded w/h extent (24 + 2*2)
#define NZ    5            // z-slab depth (K)
#define WIN   125          // 5*5*5 window
#define GROUPS    (DCUBE * DCUBE * 25)   // (w,h,k0,k1) groups per plane = 14400
#define PLANE_OUT (DCUBE * DCUBE * WIN)  // outputs per (c,l) plane = 72000

// Launch geometry baked as constants (must match kernel_launch below).
#define NTHREADS 256
#define SPLIT    4
#define GSTRIDE  (NTHREADS * SPLIT)      // 1024

// ---- CDNA5 async global->LDS copy (ASYNCcnt path), with inline-asm fallback ----
__device__ __forceinline__ void async_ld_b32(const float* g, float* l) {
#if defined(__HIP_DEVICE_COMPILE__) && __has_builtin(__builtin_amdgcn_global_load_async_to_lds_b32)
  typedef __attribute__((address_space(1))) int gi_t;   // global int*
  typedef __attribute__((address_space(3))) int li_t;   // LDS int*
  __builtin_amdgcn_global_load_async_to_lds_b32(
      (gi_t*)(unsigned long long)(uintptr_t)g,          // global src
      (li_t*)(unsigned)(uintptr_t)l,                    // LDS dst (low 32 bits of flat = LDS offset)
      0, 0);
#else
  unsigned loff = (unsigned)(uintptr_t)l;               // LDS byte offset
  asm volatile("global_load_async_to_lds_b32 %0, %1, off"
               :: "v"(loff), "v"(g) : "memory");
#endif
}

__device__ __forceinline__ void wait_async0() {
#if defined(__HIP_DEVICE_COMPILE__) && __has_builtin(__builtin_amdgcn_s_wait_asynccnt)
  __builtin_amdgcn_s_wait_asynccnt(0);
#else
  asm volatile("s_wait_asynccnt 0" ::: "memory");
#endif
}

// One block (y-split) per (c,l) plane. LDS holds the padded 5x28x28 halo slab.
__global__ __launch_bounds__(NTHREADS)
void SelfCorrelationComputation_67714454389070_kernel(const float* __restrict__ q,
                                                      float* __restrict__ out) {
  __shared__ float tile[NZ * PD * PD];   // 3920 floats = 15.7 KB

  const int plane = blockIdx.x;          // c*24 + l
  const int c     = plane / DCUBE;
  const int l     = plane - c * DCUBE;
  const int tid   = threadIdx.x;

  // Phase 1: zero-fill LDS (implements the zero padding at all halo cells).
#pragma unroll
  for (int i = 0; i < (NZ * PD * PD + NTHREADS - 1) / NTHREADS; ++i) {
    const int idx = tid + i * NTHREADS;
    if (idx < NZ * PD * PD) tile[idx] = 0.0f;
  }
  __syncthreads();

  // Phase 2: async-stage the up-to-5 valid z planes (24x24 interior each) into LDS.
  const float* qc = q + (size_t)c * (DCUBE * DCUBE * DCUBE);
  for (int zi = 0; zi < NZ; ++zi) {
    const int z = l + zi - 2;
    if (z < 0 || z >= DCUBE) continue;   // stays zero
    const float* qz = qc + z * (DCUBE * DCUBE);
    for (int e = tid; e < DCUBE * DCUBE; e += NTHREADS) {
      const int w = e / DCUBE;
      const int h = e - w * DCUBE;
      async_ld_b32(qz + e, &tile[(zi * PD + (w + 2)) * PD + (h + 2)]);
    }
  }
  wait_async0();       // this wave's async copies landed in LDS
  __syncthreads();     // all waves' copies visible

  // Phase 3: stream the 72000 contiguous outputs of this plane.
  // One group g = (voxel, k0, k1) -> 5 consecutive outputs (k2 = 0..4).
  float* outp = out + (size_t)plane * PLANE_OUT;
  const int g0 = blockIdx.y * NTHREADS + tid;
  for (int g = g0; g < GROUPS; g += GSTRIDE) {
    const int voxel = g / 25;
    const int r     = g - voxel * 25;    // k0*5 + k1
    const int k0    = r / 5;
    const int k1    = r - k0 * 5;
    const int w     = voxel / DCUBE;
    const int h     = voxel - w * DCUBE;

    const float center = tile[(2 * PD + (w + 2)) * PD + (h + 2)];
    const int   nb     = (k0 * PD + (w + k1)) * PD + h;  // +k2 gives padded col h+k2
    float* o = outp + g * 5;
#pragma unroll
    for (int k2 = 0; k2 < 5; ++k2) {
      const float v = center * tile[nb + k2];
      o[k2] = fmaxf(v, 0.0f);
    }
  }
}

extern "C" void kernel_launch(void* const* d_in, const int* in_sizes, int n_in,
                              void* d_out, int out_size, void* d_ws, size_t ws_size,
                              hipStream_t stream) {
  const float* q  = (const float*)d_in[0];
  float*       ot = (float*)d_out;
  const int c_total = in_sizes[0] / (DCUBE * DCUBE * DCUBE);   // 32
  dim3 grid(c_total * DCUBE, SPLIT, 1);   // 768 planes x 4 splits = 3072 blocks
  dim3 block(NTHREADS, 1, 1);
  SelfCorrelationComputation_67714454389070_kernel<<<grid, block, 0, stream>>>(q, ot);
}